// SpatialCrossAttention_21741124453153
// MI455X (gfx1250) — compile-verified
//
#include <hip/hip_runtime.h>
#include <hip/hip_bf16.h>
#include <math.h>

typedef float v2f __attribute__((ext_vector_type(2)));
typedef float v8f __attribute__((ext_vector_type(8)));

#define NQ     10000
#define NQPAD  10048   // 157 blocks * 64 rows
#define CCH    256
#define NCAM   6
#define ZB     4
#define HFEAT  64
#define WFEAT  176
#define NCOMBO 24      // NCAM * ZB

// ---------------------------------------------------------------------------
// Kernel 1: attn[q, t] = sigmoid(query[q,:] . W_attn[:,t] + b_attn[t])
// WMMA f32 16x16x4: M=10000 (625 tiles), N=24 padded to 32 (2 tiles), K=256.
// W_attn staged zero-padded [256][32] in LDS. One wave per M-tile.
// ---------------------------------------------------------------------------
__global__ void attn_wmma_kernel(const float* __restrict__ query,   // [NQ,256]
                                 const float* __restrict__ W_attn,  // [256,24]
                                 const float* __restrict__ b_attn,  // [24]
                                 float* __restrict__ attnw) {       // [NQ,24]
  __shared__ float lds[CCH * 32];   // 32 KB, W_attn padded to 32 cols

  const int tid = threadIdx.x;
  for (int i = tid; i < CCH * 32; i += 256) {
    const int k = i >> 5;
    const int n = i & 31;
    lds[i] = (n < NCOMBO) ? W_attn[k * NCOMBO + n] : 0.f;
  }
  __syncthreads();

  const int wid  = tid >> 5;
  const int lane = tid & 31;
  const int l16  = lane & 15;
  const int half = lane >> 4;
  const int tile = blockIdx.x * 8 + wid;        // M tile index
  const int m0   = tile * 16;
  if (m0 >= NQ) return;                          // whole-wave uniform exit
  const int row  = m0 + l16;
  const int rowc = row < NQ ? row : NQ - 1;      // branch-free clamp
  const float* qrow = query + (size_t)rowc * CCH;

  v8f c0 = {};
  v8f c1 = {};
#pragma unroll 8
  for (int k = 0; k < CCH; k += 4) {
    const int kk = k + 2 * half;                 // lanes 0-15: K pair {k,k+1}
    const v2f a = *(const v2f*)(qrow + kk);
    v2f b0, b1;
    b0.x = lds[kk * 32 + l16];
    b0.y = lds[(kk + 1) * 32 + l16];
    b1.x = lds[kk * 32 + 16 + l16];
    b1.y = lds[(kk + 1) * 32 + 16 + l16];
    c0 = __builtin_amdgcn_wmma_f32_16x16x4_f32(false, a, false, b0, (short)0, c0, false, false);
    c1 = __builtin_amdgcn_wmma_f32_16x16x4_f32(false, a, false, b1, (short)0, c1, false, false);
  }

  // D(m,n): VGPR r, lane l16 -> m = m0 + r + 8*half, n = n0 + l16
#pragma unroll
  for (int t = 0; t < 2; ++t) {
    const int col = t * 16 + l16;
    if (col < NCOMBO) {
      const float bb = b_attn[col];
      const v8f& c = t ? c1 : c0;
#pragma unroll
      for (int r = 0; r < 8; ++r) {
        const int m = m0 + r + 8 * half;
        if (m < NQ) {
          const float s = c[r] + bb;
          attnw[m * NCOMBO + col] = 1.f / (1.f + __expf(-s));
        }
      }
    }
  }
}

// ---------------------------------------------------------------------------
// Kernel 2: fused projection + mask + bilinear sample + weighted accumulate.
// One block per (padded) query row; 256 threads = one channel each.
// Padding rows (q >= NQ) just write zeros so the GEMM can load unguarded.
// ---------------------------------------------------------------------------
__global__ void sample_kernel(const float* __restrict__ img,        // [6,256,64,176]
                              const float* __restrict__ lidar2img,  // [6,4,4]
                              const float* __restrict__ attnw,      // [NQ,24]
                              float* __restrict__ acc) {            // [NQPAD,256]
  __shared__ int   s_x0[NCOMBO], s_y0[NCOMBO];
  __shared__ float s_wx[NCOMBO], s_wy[NCOMBO], s_w[NCOMBO];

  const int q = blockIdx.x;
  const int c = threadIdx.x;
  if (q >= NQ) {                       // zero the GEMM padding rows
    acc[(size_t)q * CCH + c] = 0.f;
    return;
  }

  if (c < NCOMBO) {
    const int t = c;
    const int n = t >> 2;
    const int z = t & 3;
    const int iy = q / 100;
    const int ix = q - iy * 100;
    const float X  = ((ix + 0.5f) * 0.01f) * 102.4f - 51.2f;
    const float Y  = ((iy + 0.5f) * 0.01f) * 102.4f - 51.2f;
    const float Zc = ((z  + 0.5f) * 0.25f) * 8.0f   - 5.0f;
    const float* L = lidar2img + n * 16;
    const float c0 = fmaf(L[0], X, fmaf(L[1], Y, fmaf(L[2],  Zc, L[3])));
    const float c1 = fmaf(L[4], X, fmaf(L[5], Y, fmaf(L[6],  Zc, L[7])));
    const float c2 = fmaf(L[8], X, fmaf(L[9], Y, fmaf(L[10], Zc, L[11])));
    const bool valid = c2 > 1e-5f;
    const float d = fmaxf(c2, 1e-5f);
    const float u = c0 / d;
    const float v = c1 / d;
    const float gx = u * (1.0f / WFEAT);
    const float gy = v * (1.0f / HFEAT);
    const bool m = valid && (gx > 0.f) && (gx < 1.f) && (gy > 0.f) && (gy < 1.f);
    const float px = u - 0.5f;
    const float py = v - 0.5f;
    const float x0f = floorf(px);
    const float y0f = floorf(py);
    s_x0[t] = (int)x0f;
    s_y0[t] = (int)y0f;
    s_wx[t] = px - x0f;
    s_wy[t] = py - y0f;
    s_w[t]  = m ? attnw[q * NCOMBO + t] : 0.f;
  }
  __syncthreads();

  float a = 0.f;
#pragma unroll 1
  for (int t = 0; t < NCOMBO; ++t) {
    const float w = s_w[t];
    if (w == 0.f) continue;            // block-uniform branch
    const int n = t >> 2;
    const float* f = img + ((size_t)(n * CCH + c)) * (HFEAT * WFEAT);
    const int x0 = s_x0[t];
    const int y0 = s_y0[t];
    const float wx = s_wx[t];
    const float wy = s_wy[t];
    float v00 = 0.f, v01 = 0.f, v10 = 0.f, v11 = 0.f;
    const bool xi0 = (unsigned)x0       < (unsigned)WFEAT;
    const bool xi1 = (unsigned)(x0 + 1) < (unsigned)WFEAT;
    if ((unsigned)y0 < (unsigned)HFEAT) {
      const float* r = f + y0 * WFEAT;
      if (xi0) v00 = r[x0];
      if (xi1) v01 = r[x0 + 1];
    }
    if ((unsigned)(y0 + 1) < (unsigned)HFEAT) {
      const float* r = f + (y0 + 1) * WFEAT;
      if (xi0) v10 = r[x0];
      if (xi1) v11 = r[x0 + 1];
    }
    const float top = fmaf(v01 - v00, wx, v00);
    const float bot = fmaf(v11 - v10, wx, v10);
    a = fmaf(w, fmaf(bot - top, wy, top), a);
  }
  acc[(size_t)q * CCH + c] = a;
}

// ---------------------------------------------------------------------------
// async DMA of one 64x256 f32 slab of W_out into an LDS buffer (16 B / lane).
// ---------------------------------------------------------------------------
__device__ __forceinline__ void issue_slab_async(unsigned lds_byte_base,
                                                 const float* __restrict__ gsrc,
                                                 int tid) {
#pragma unroll
  for (int it = 0; it < 16; ++it) {
    const int e = (tid + it * 256) * 4;            // float index
    const unsigned loff = lds_byte_base + e * 4;   // LDS byte offset
    const float* ga = gsrc + e;
    asm volatile("global_load_async_to_lds_b128 %0, %1, off"
                 :: "v"(loff), "v"(ga) : "memory");
  }
}

// ---------------------------------------------------------------------------
// Kernel 3: out = acc @ W_out + b_out + query   (fp32 WMMA 16x16x4)
// Block = 256 threads (8 waves) computing a 64x256 output stripe.
// W_out streamed through a 2 x 64 KB LDS double buffer with
// GLOBAL_LOAD_ASYNC_TO_LDS_B128; s_wait_asynccnt 16 overlaps slab s+1 DMA
// with slab s WMMA compute (async loads complete in order).
// ---------------------------------------------------------------------------
__global__ void out_gemm_kernel(const float* __restrict__ acc,     // [NQPAD,256]
                                const float* __restrict__ W_out,   // [256,256]
                                const float* __restrict__ b_out,   // [256]
                                const float* __restrict__ query,   // [NQ,256]
                                float* __restrict__ out) {         // [NQ,256]
  __shared__ float lds[2][64 * CCH];       // 128 KB double buffer

  const int tid  = threadIdx.x;
  const int wid  = tid >> 5;               // 0..7
  const int lane = tid & 31;
  const int l16  = lane & 15;
  const int half = lane >> 4;
  const int tm   = wid >> 1;               // 0..3  (M tile in stripe)
  const int tn0  = (wid & 1) * 8;          // 0 or 8 (first N tile)
  const int m0   = blockIdx.x * 64 + tm * 16;
  const int row  = m0 + l16;
  const float* arow = acc + (size_t)row * CCH;
  const unsigned lb0 = (unsigned)(uintptr_t)(&lds[0][0]);
  const unsigned lb1 = (unsigned)(uintptr_t)(&lds[1][0]);

  v8f cacc[8] = {};

  issue_slab_async(lb0, W_out, tid);       // prologue: slab 0 -> buf 0

  for (int s = 0; s < 4; ++s) {
    const int b  = s & 1;
    const int k0 = s * 64;
    if (s + 1 < 4) {
      __syncthreads();                     // all waves done reading buf[1-b]
      issue_slab_async(b ? lb0 : lb1, W_out + (k0 + 64) * CCH, tid);
      // async loads retire in order: <=16 outstanding => slab s has landed
      asm volatile("s_wait_asynccnt 0x10" ::: "memory");
      __builtin_prefetch(arow + k0 + 64, 0, 1);   // warm next A slab
    } else {
      asm volatile("s_wait_asynccnt 0x0" ::: "memory");
    }
    __syncthreads();                       // slab s visible to every wave

    const float* __restrict__ wb = &lds[b][0];
#pragma unroll 8
    for (int k = 0; k < 64; k += 4) {
      const int kk = k + 2 * half;         // lanes 0-15: K={k,k+1}; 16-31: K={k+2,k+3}
      const v2f afrag = *(const v2f*)(arow + k0 + kk);
#pragma unroll
      for (int t = 0; t < 8; ++t) {
        const int n0 = (tn0 + t) << 4;
        v2f bfrag;
        bfrag.x = wb[kk * CCH + n0 + l16];
        bfrag.y = wb[(kk + 1) * CCH + n0 + l16];
        cacc[t] = __builtin_amdgcn_wmma_f32_16x16x4_f32(
            false, afrag, false, bfrag, (short)0, cacc[t], false, false);
      }
    }
  }

  // epilogue: D(m,n) at VGPR r, lane l16: m = m0 + r + 8*half, n = n0 + l16
#pragma unroll
  for (int t = 0; t < 8; ++t) {
    const int col = ((tn0 + t) << 4) + l16;
    const float bb = b_out[col];
#pragma unroll
    for (int r = 0; r < 8; ++r) {
      const int m = m0 + r + 8 * half;
      if (m < NQ)
        out[(size_t)m * CCH + col] = cacc[t][r] + bb + query[(size_t)m * CCH + col];
    }
  }
}

// ---------------------------------------------------------------------------
extern "C" void kernel_launch(void* const* d_in, const int* in_sizes, int n_in,
                              void* d_out, int out_size, void* d_ws, size_t ws_size,
                              hipStream_t stream) {
  const float* query  = (const float*)d_in[0];   // [1,10000,256]
  const float* img    = (const float*)d_in[1];   // [1,6,256,64,176]
  const float* l2i    = (const float*)d_in[2];   // [1,6,4,4]
  const float* W_attn = (const float*)d_in[3];   // [256,24]
  const float* b_attn = (const float*)d_in[4];   // [24]
  const float* W_out  = (const float*)d_in[5];   // [256,256]
  const float* b_out  = (const float*)d_in[6];   // [256]
  float* out = (float*)d_out;                    // [1,10000,256]

  float* acc   = (float*)d_ws;                   // NQPAD*256 floats (padded)
  float* attnw = acc + (size_t)NQPAD * CCH;      // NQ*24 floats

  attn_wmma_kernel<<<(NQ / 16 + 7) / 8, 256, 0, stream>>>(query, W_attn, b_attn, attnw);
  sample_kernel<<<NQPAD, 256, 0, stream>>>(img, l2i, attnw, acc);
  out_gemm_kernel<<<NQPAD / 64, 256, 0, stream>>>(acc, W_out, b_out, query, out);
}